// Threshold_weights8_52699248721955
// MI455X (gfx1250) — compile-verified
//
#include <hip/hip_runtime.h>
#include <math.h>

#define NROWS 16384
#define NCLS  1000
#define NMODELS 9
#define WAVES_PER_BLOCK 8
#define MAXSLOTS 64          // 64 cacheline-spaced atomic-max slots
#define SLOT_STRIDE_U32 16   // 64 bytes apart

// gfx1250 async global->LDS path (builtin confirmed present by round-1 probe).
#if defined(__has_builtin)
#  if __has_builtin(__builtin_amdgcn_global_load_async_to_lds_b128) && \
      __has_builtin(__builtin_amdgcn_s_wait_asynccnt)
#    define USE_ASYNC_LDS 1
#  endif
#endif

struct Ptrs { const float* p[NMODELS]; };

// Monotonic float -> uint encoding so atomicMax(uint) == max(float).
__device__ __forceinline__ unsigned encf(float f) {
  unsigned u = __float_as_uint(f);
  return (u & 0x80000000u) ? ~u : (u | 0x80000000u);
}
__device__ __forceinline__ float decf(unsigned u) {
  return (u & 0x80000000u) ? __uint_as_float(u & 0x7FFFFFFFu)
                           : __uint_as_float(~u);
}

// Branchless running top-2 (keeps duplicates, like lax.top_k).
__device__ __forceinline__ void top2_update(float v, float& m1, float& m2) {
  float hi = fmaxf(v, m1);
  float lo = fminf(v, m1);
  m1 = hi;
  m2 = fmaxf(m2, lo);
}

#ifdef USE_ASYNC_LDS
// Exact parameter types per hipcc diagnostic: v4i in AS1 (global) / AS3 (LDS).
typedef int v4i __attribute__((vector_size(4 * sizeof(int))));
typedef __attribute__((address_space(1))) v4i* g_v4i_ptr;
typedef __attribute__((address_space(3))) v4i* l_v4i_ptr;
#endif

__global__ __launch_bounds__(256)
void margins_kernel(Ptrs ptrs, const int* __restrict__ targets,
                    float* __restrict__ out, unsigned* __restrict__ slots) {
  const int lane = threadIdx.x & 31;
  const int wave = threadIdx.x >> 5;
  const int w    = blockIdx.x * WAVES_PER_BLOCK + wave;   // exact grid: no bounds check
  const int row  = w / NMODELS;
  const int mat  = w - row * NMODELS;

  const float* __restrict__ x = ptrs.p[mat] + (size_t)row * NCLS;

  float m1 = -__builtin_huge_valf();
  float m2 = -__builtin_huge_valf();
  float tgt;
  const int t = targets[row];

#ifdef USE_ASYNC_LDS
  // Stage this wave's row (4000 B) into LDS with async b128 copies,
  // then reduce out of LDS. ASYNCcnt is per-wave -> no barrier needed.
  __shared__ float lds[WAVES_PER_BLOCK * NCLS];
  float* myl = lds + wave * NCLS;
#pragma unroll
  for (int i = 0; i < 8; ++i) {
    int c = lane + (i << 5);
    if (c < NCLS / 4) {
      __builtin_amdgcn_global_load_async_to_lds_b128(
          (g_v4i_ptr)(x + 4 * c), (l_v4i_ptr)(myl + 4 * c), 0, 0);
    }
  }
  __builtin_amdgcn_s_wait_asynccnt(0);
  const float4* __restrict__ s4 = (const float4*)myl;
#pragma unroll
  for (int i = 0; i < 8; ++i) {
    int c = lane + (i << 5);
    if (c < NCLS / 4) {
      float4 v = s4[c];
      top2_update(v.x, m1, m2);
      top2_update(v.y, m1, m2);
      top2_update(v.z, m1, m2);
      top2_update(v.w, m1, m2);
    }
  }
  tgt = myl[t];
#else
  // Row stride is 4000 B = 250 * 16 B -> every row is float4-aligned.
  const float4* __restrict__ x4 = (const float4*)x;
#pragma unroll
  for (int i = 0; i < 8; ++i) {
    int c = lane + (i << 5);
    if (c < NCLS / 4) {
      float4 v = x4[c];
      top2_update(v.x, m1, m2);
      top2_update(v.y, m1, m2);
      top2_update(v.z, m1, m2);
      top2_update(v.w, m1, m2);
    }
  }
  tgt = x[t];   // broadcast load, single cacheline
#endif

  // Wave-level top-2 merge across 32 lanes.
#pragma unroll
  for (int off = 16; off > 0; off >>= 1) {
    float o1 = __shfl_xor(m1, off, 32);
    float o2 = __shfl_xor(m2, off, 32);
    float hi = fmaxf(m1, o1);
    float lo = fminf(m1, o1);
    m2 = fmaxf(lo, fmaxf(m2, o2));
    m1 = hi;
  }

  if (lane == 0) {
    float margin = (tgt == m1) ? (m1 - m2) : 0.0f;
    out[1 + (size_t)row * NMODELS + mat] = margin;      // raw margin; kernel2 softmaxes in place
    if (mat < 8) {
      atomicMax(&slots[(w & (MAXSLOTS - 1)) * SLOT_STRIDE_U32], encf(m1));
    }
  }
}

__global__ __launch_bounds__(256)
void softmax_finalize_kernel(float* __restrict__ out,
                             const unsigned* __restrict__ slots) {
  const int r = blockIdx.x * blockDim.x + threadIdx.x;
  if (r >= NROWS) return;

  float* base = out + 1 + (size_t)r * NMODELS;
  float v[NMODELS];
  float mx = -__builtin_huge_valf();
#pragma unroll
  for (int j = 0; j < NMODELS; ++j) {
    v[j] = base[j] * 0.5f;               // margins / T, T = 2.0
    mx = fmaxf(mx, v[j]);
  }
  float s = 0.0f;
#pragma unroll
  for (int j = 0; j < NMODELS; ++j) {
    v[j] = expf(v[j] - mx);
    s += v[j];
  }
#pragma unroll
  for (int j = 0; j < NMODELS; ++j) base[j] = v[j] / s;

  if (r == 0) {
    unsigned u = 0;
#pragma unroll
    for (int i = 0; i < MAXSLOTS; ++i) {
      unsigned q = slots[i * SLOT_STRIDE_U32];
      u = (q > u) ? q : u;
    }
    out[0] = decf(u);                    // global max over outputs1..8
  }
}

extern "C" void kernel_launch(void* const* d_in, const int* in_sizes, int n_in,
                              void* d_out, int out_size, void* d_ws, size_t ws_size,
                              hipStream_t stream) {
  (void)in_sizes; (void)n_in; (void)out_size; (void)ws_size;

  Ptrs ptrs;
  for (int i = 0; i < NMODELS; ++i) ptrs.p[i] = (const float*)d_in[i];  // outputs1..8, mimic
  const int* targets = (const int*)d_in[9];
  float* out = (float*)d_out;
  unsigned* slots = (unsigned*)d_ws;     // 64 slots * 64 B = 4 KB of scratch

  // Zero the atomic-max slots every call (graph-capture-safe).
  (void)hipMemsetAsync(d_ws, 0, MAXSLOTS * SLOT_STRIDE_U32 * sizeof(unsigned), stream);

  const int total_waves = NROWS * NMODELS;               // 147456
  const int blocks = total_waves / WAVES_PER_BLOCK;      // 18432, exact

  margins_kernel<<<blocks, 256, 0, stream>>>(ptrs, targets, out, slots);
  softmax_finalize_kernel<<<(NROWS + 255) / 256, 256, 0, stream>>>(out, slots);
}